// PointVoteNMS_84164179132856
// MI455X (gfx1250) — compile-verified
//
#include <hip/hip_runtime.h>
#include <hip/hip_bf16.h>

#define INF __builtin_inff()

#define NPTS        300000
#define NCLASSES    20
#define KINST       100
#define KPAD        112      // 7 tiles of 16
#define NTILES      7
#define NINST       101      // instance ids 0..100
#define NCLSVOTE    100
#define NBLK        512      // partial-reduction blocks for NMS scan

typedef __attribute__((ext_vector_type(2))) float v2f;
typedef __attribute__((ext_vector_type(8))) float v8f;

struct Partial {
    float v;      // block max of (new) scores
    int   idx;    // first index achieving it
    int   votes;  // matched count
    float sx, sy, sz;
};

// ---------------------------------------------------------------- prep ------
__global__ void k_prep(const float* __restrict__ xyz,
                       const float* __restrict__ sem,
                       const float* __restrict__ off,
                       const float* __restrict__ hmap,
                       float4* __restrict__ shifted,
                       int* __restrict__ pan, int n) {
    int i = blockIdx.x * blockDim.x + threadIdx.x;
    if (i >= n) return;
    const float* s = sem + (long)i * NCLASSES;
    float best = s[0]; int bi = 0;
#pragma unroll
    for (int j = 1; j < NCLASSES; ++j) {
        float v = s[j];
        bool take = v > best;                 // first-index tie-break
        best = take ? v : best;
        bi   = take ? j : bi;
    }
    pan[i] = bi;
    float x = xyz[3 * i + 0] + off[3 * i + 0];
    float y = xyz[3 * i + 1] + off[3 * i + 1];
    float z = xyz[3 * i + 2] + off[3 * i + 2];
    bool fg = (bi >= 1) && (bi <= 8);
    float h = hmap[i];
    float sc = (fg && h > 0.15f) ? h : -INF;
    shifted[i] = make_float4(x, y, z, sc);
}

// ------------------------------------------------------------- init ---------
__global__ void k_init(float4* __restrict__ cB, float4* __restrict__ state,
                       int* __restrict__ counts) {
    int i = blockIdx.x * blockDim.x + threadIdx.x;
    if (i < KPAD) cB[i] = make_float4(0.f, 0.f, 0.f, 1e30f);
    if (i == 0)   state[0] = make_float4(3e18f, 3e18f, 3e18f, -INF); // sentinel
    if (i < NCLSVOTE * NINST) counts[i] = 0;
}

// ----------------------------------------------- NMS scan (fused round) -----
__global__ void k_scan(float4* __restrict__ shifted,
                       const float4* __restrict__ state,
                       Partial* __restrict__ part, int n) {
    __shared__ float sv[256];  __shared__ int sidx[256]; __shared__ int svt[256];
    __shared__ float ssx[256]; __shared__ float ssy[256]; __shared__ float ssz[256];
    int t = threadIdx.x;
    float4 st = state[0];
    int chunk = (n + NBLK - 1) / NBLK;
    int start = blockIdx.x * chunk;
    int end   = min(start + chunk, n);
    float bestv = -INF; int besti = 0x7fffffff;
    int votes = 0; float sx = 0.f, sy = 0.f, sz = 0.f;
    for (int i = start + t; i < end; i += 256) {
        float4 p = shifted[i];
        float dx = p.x - st.x, dy = p.y - st.y, dz = p.z - st.z;
        float d2 = dx * dx + dy * dy + dz * dz;
        bool fin = (p.w != -INF);
        bool match = (d2 < 1.0f) && fin;
        votes += match ? 1 : 0;
        sx += match ? p.x : 0.0f;
        sy += match ? p.y : 0.0f;
        sz += match ? p.z : 0.0f;
        if (match) ((float*)shifted)[4 * i + 3] = -INF;   // suppress (score only)
        float ns = match ? -INF : p.w;
        bool take = (ns > bestv) || (ns == bestv && i < besti);
        bestv = take ? ns : bestv;
        besti = take ? i  : besti;
    }
    sv[t] = bestv; sidx[t] = besti; svt[t] = votes; ssx[t] = sx; ssy[t] = sy; ssz[t] = sz;
    __syncthreads();
    for (int s = 128; s > 0; s >>= 1) {
        if (t < s) {
            float ov = sv[t + s]; int oi = sidx[t + s];
            bool take = (ov > sv[t]) || (ov == sv[t] && oi < sidx[t]);
            sv[t]   = take ? ov : sv[t];
            sidx[t] = take ? oi : sidx[t];
            svt[t] += svt[t + s]; ssx[t] += ssx[t + s]; ssy[t] += ssy[t + s]; ssz[t] += ssz[t + s];
        }
        __syncthreads();
    }
    if (t == 0) {
        Partial p; p.v = sv[0]; p.idx = sidx[0]; p.votes = svt[0];
        p.sx = ssx[0]; p.sy = ssy[0]; p.sz = ssz[0];
        part[blockIdx.x] = p;
    }
}

// ------------------------------------------------ NMS finalize (1 block) ----
__global__ void k_final(const Partial* __restrict__ part,
                        const float4* __restrict__ shifted,
                        float4* __restrict__ state,
                        float4* __restrict__ cB,
                        float* __restrict__ outCenters, int round) {
    __shared__ float sv[NBLK];  __shared__ int sidx[NBLK]; __shared__ int svt[NBLK];
    __shared__ float ssx[NBLK]; __shared__ float ssy[NBLK]; __shared__ float ssz[NBLK];
    int t = threadIdx.x;
    Partial p = part[t];
    sv[t] = p.v; sidx[t] = p.idx; svt[t] = p.votes; ssx[t] = p.sx; ssy[t] = p.sy; ssz[t] = p.sz;
    __syncthreads();
    for (int s = NBLK / 2; s > 0; s >>= 1) {
        if (t < s) {
            float ov = sv[t + s]; int oi = sidx[t + s];
            bool take = (ov > sv[t]) || (ov == sv[t] && oi < sidx[t]);
            sv[t]   = take ? ov : sv[t];
            sidx[t] = take ? oi : sidx[t];
            svt[t] += svt[t + s]; ssx[t] += ssx[t + s]; ssy[t] += ssy[t + s]; ssz[t] += ssz[t + s];
        }
        __syncthreads();
    }
    if (t == 0) {
        if (round >= 0) {
            float s_sel = state[0].w;                       // score at selection time
            int votes = svt[0];
            bool keep = (s_sel != -INF) && (votes >= 25);
            float denom = fmaxf((float)votes, 1.0f);
            float cx = keep ? ssx[0] / denom : 0.0f;
            float cy = keep ? ssy[0] / denom : 0.0f;
            float cz = keep ? ssz[0] / denom : 0.0f;
            outCenters[3 * round + 0] = cx;
            outCenters[3 * round + 1] = cy;
            outCenters[3 * round + 2] = cz;
            cB[round] = keep ? make_float4(cx, cy, cz, cx * cx + cy * cy + cz * cz)
                             : make_float4(0.f, 0.f, 0.f, 1e30f);
        }
        int idx = sidx[0];
        float4 q = shifted[idx];
        state[0] = make_float4(q.x, q.y, q.z, sv[0]);       // next round's candidate
    }
}

// --------------------------------------- WMMA nearest-center assignment -----
// S[m][n] = |c_n|^2 - 2 p_m . c_n  via V_WMMA_F32_16X16X4_F32.
// A row m = (-2x, -2y, -2z, 1); B col n = (cx, cy, cz, |c|^2 or 1e30 mask).
// ds_swizzle xor-butterfly patterns (group-of-32: offset = (xor<<10)|0x1f)
#define SWZ1 0x041f
#define SWZ2 0x081f
#define SWZ4 0x101f
#define SWZ8 0x201f

__global__ void k_assign(const float4* __restrict__ shifted,
                         const float4* __restrict__ cB,
                         int* __restrict__ inst, int ngroups) {
    int t = threadIdx.x;
    int lane = t & 31;
    int wave = t >> 5;
    int m = lane & 15;
    bool hi = lane >= 16;               // lanes 16..31 carry K=2,3 halves
    // B operands are group-invariant: preload all 7 tiles into registers.
    v2f b[NTILES];
#pragma unroll
    for (int tile = 0; tile < NTILES; ++tile) {
        float4 c = cB[tile * 16 + m];
        b[tile].x = hi ? c.z : c.x;
        b[tile].y = hi ? c.w : c.y;
    }
    int gstride = gridDim.x * 8;        // 8 waves per 256-thread block
    for (int g = blockIdx.x * 8 + wave; g < ngroups; g += gstride) {
        int base = g * 16;
        float4 p = shifted[base + m];
        v2f a;
        a.x = hi ? (-2.0f * p.z) : (-2.0f * p.x);
        a.y = hi ? 1.0f          : (-2.0f * p.y);
        float bv[8]; int bk[8];
#pragma unroll
        for (int r = 0; r < 8; ++r) { bv[r] = 1e38f; bk[r] = 0x7fffffff; }
        // Per-lane reduction across tiles (branchless; earliest tile wins ties)
#pragma unroll
        for (int tile = 0; tile < NTILES; ++tile) {
            v8f acc = {};
            acc = __builtin_amdgcn_wmma_f32_16x16x4_f32(
                false, a, false, b[tile], (short)0, acc, false, false);
            int kk = tile * 16 + m;
#pragma unroll
            for (int r = 0; r < 8; ++r) {
                float v = acc[r];
                bool take = v < bv[r];
                bv[r] = take ? v  : bv[r];
                bk[r] = take ? kk : bk[r];
            }
        }
        // One cross-lane reduction per row: min value, then min index among
        // lanes achieving it (== first-index argmin). Fully branchless.
#pragma unroll
        for (int r = 0; r < 8; ++r) {
            float v = bv[r];
            v = fminf(v, __int_as_float(__builtin_amdgcn_ds_swizzle(__float_as_int(v), SWZ1)));
            v = fminf(v, __int_as_float(__builtin_amdgcn_ds_swizzle(__float_as_int(v), SWZ2)));
            v = fminf(v, __int_as_float(__builtin_amdgcn_ds_swizzle(__float_as_int(v), SWZ4)));
            v = fminf(v, __int_as_float(__builtin_amdgcn_ds_swizzle(__float_as_int(v), SWZ8)));
            int cand = (bv[r] == v) ? bk[r] : 0x7fffffff;
            cand = min(cand, __builtin_amdgcn_ds_swizzle(cand, SWZ1));
            cand = min(cand, __builtin_amdgcn_ds_swizzle(cand, SWZ2));
            cand = min(cand, __builtin_amdgcn_ds_swizzle(cand, SWZ4));
            cand = min(cand, __builtin_amdgcn_ds_swizzle(cand, SWZ8));
            bk[r] = cand;
        }
        if ((lane & 15) == 0) {          // lane 0 -> rows 0..7, lane 16 -> rows 8..15
            int rb = base + (hi ? 8 : 0);
#pragma unroll
            for (int r = 0; r < 8; ++r) inst[rb + r] = bk[r] + 1;
        }
    }
}

// ------------------------------------------------------- histogram ----------
__global__ void k_hist(const int* __restrict__ pan, const int* __restrict__ inst,
                       int* __restrict__ counts, int n) {
    int i = blockIdx.x * blockDim.x + threadIdx.x;
    if (i >= n) return;
    int p = pan[i];
    if (p >= 1 && p <= 8) atomicAdd(&counts[p * NINST + inst[i]], 1);
}

__global__ void k_semvote(const int* __restrict__ counts, int* __restrict__ semOf) {
    int j = blockIdx.x * blockDim.x + threadIdx.x;
    if (j >= NINST) return;
    int best = counts[j]; int bs = 0;        // sem = 0 row
    for (int s = 1; s < NCLSVOTE; ++s) {
        int v = counts[s * NINST + j];
        bool take = v > best;                // first-index tie-break
        best = take ? v : best;
        bs   = take ? s : bs;
    }
    semOf[j] = bs;
}

__global__ void k_out(const int* __restrict__ pan, const int* __restrict__ inst,
                      const int* __restrict__ semOf, int* __restrict__ outI, int n) {
    int i = blockIdx.x * blockDim.x + threadIdx.x;
    if (i >= n) return;
    int p = pan[i];
    int id = inst[i];
    bool fg = (p >= 1 && p <= 8);
    outI[i] = fg ? (semOf[id] + (id << 16)) : p;
}

// ---------------------------------------------------------------- launch ----
extern "C" void kernel_launch(void* const* d_in, const int* in_sizes, int n_in,
                              void* d_out, int out_size, void* d_ws, size_t ws_size,
                              hipStream_t stream) {
    (void)in_sizes; (void)n_in; (void)out_size; (void)ws_size;
    const float* xyz  = (const float*)d_in[0];
    const float* sem  = (const float*)d_in[1];
    const float* offs = (const float*)d_in[2];
    const float* hmap = (const float*)d_in[3];

    char* ws = (char*)d_ws;
    float4* shifted = (float4*)ws;                                  // N*16B
    int*    pan     = (int*)(ws + (size_t)16 * NPTS);               // N*4B
    int*    inst    = (int*)(ws + (size_t)20 * NPTS);               // N*4B
    float4* cB      = (float4*)(ws + (size_t)24 * NPTS);            // KPAD*16B
    float4* state   = cB + KPAD;                                    // 16B
    Partial* part   = (Partial*)((char*)(state + 1));               // NBLK*24B
    int*    counts  = (int*)((char*)part + NBLK * sizeof(Partial)); // 100*101*4B
    int*    semOf   = counts + NCLSVOTE * NINST;                    // 101*4B

    float* outC = (float*)d_out;            // centers: 100 x 3 floats
    int*   outI = (int*)(outC + 3 * KINST); // panoptic: N int32

    k_prep<<<(NPTS + 255) / 256, 256, 0, stream>>>(xyz, sem, offs, hmap, shifted, pan, NPTS);
    k_init<<<(NCLSVOTE * NINST + 255) / 256, 256, 0, stream>>>(cB, state, counts);

    // Bootstrap pass (sentinel center -> no suppression) selects round-0
    // candidate; then 100 real rounds. round = j-1 is finalized at step j.
    for (int j = 0; j <= KINST; ++j) {
        k_scan<<<NBLK, 256, 0, stream>>>(shifted, state, part, NPTS);
        k_final<<<1, NBLK, 0, stream>>>(part, shifted, state, cB, outC, j - 1);
    }

    k_assign<<<240, 256, 0, stream>>>(shifted, cB, inst, NPTS / 16);
    k_hist<<<(NPTS + 255) / 256, 256, 0, stream>>>(pan, inst, counts, NPTS);
    k_semvote<<<1, 128, 0, stream>>>(counts, semOf);
    k_out<<<(NPTS + 255) / 256, 256, 0, stream>>>(pan, inst, semOf, outI, NPTS);
}